// EGNNConv_3719441678490
// MI455X (gfx1250) — compile-verified
//
#include <hip/hip_runtime.h>
#include <math.h>

// EGNN conv for MI455X (gfx1250): fp32 WMMA 16x16x4 path.
// D = H = 64; edge/node tiles of 16 rows per wave32.
// Weights held in LDS pair-major so each WMMA B fragment is one ds_load_b64.

#define STP 68            // stage pitch in floats (272 B: 16B aligned, conflict-free)
#define WAVES 4
#define BLOCK (WAVES * 32)

typedef __attribute__((ext_vector_type(2))) float v2f;
typedef __attribute__((ext_vector_type(8))) float v8f;

__device__ __forceinline__ float silu_f(float x) {
  return x / (1.0f + __expf(-x));
}

__device__ __forceinline__ v8f v8f_zero() {
  v8f z;
#pragma unroll
  for (int i = 0; i < 8; ++i) z[i] = 0.0f;
  return z;
}

__device__ __forceinline__ v8f wmma_f32_16x16x4(v2f a, v2f b, v8f c) {
  // (neg_a, A, neg_b, B, c_mod, C, reuse_a, reuse_b)
  return __builtin_amdgcn_wmma_f32_16x16x4_f32(false, a, false, b, (short)0, c,
                                               false, false);
}

// Cooperative load of `rows`x64 weight block into pair-major LDS layout:
//   sW2[(k>>1)*128 + n*2 + (k&1)] = W[k*64 + n]
__device__ __forceinline__ void load_w_pairmajor(const float* __restrict__ g,
                                                 float* __restrict__ sW2,
                                                 int rows, int tid) {
  for (int i = tid; i < rows * 64; i += BLOCK) {
    const int k = i >> 6;
    const int n = i & 63;
    sW2[((k >> 1) << 7) + (n << 1) + (k & 1)] = g[i];
  }
}

// C[16x64] += A[16x64] * B[64x64]
// A: LDS stage, row-major [16][STP] (8B-aligned pairs).
// B: LDS pair-major [32 pairs][64 n][2].
// acc: 4 N-tiles of the 16x16 f32 C/D fragment (8 VGPRs each).
// f32 16x4 A frag: lane<16 -> row=lane, k..k+1 ; lane>=16 -> row=lane-16, k+2..k+3.
__device__ __forceinline__ void gemm_16x64x64(const float* __restrict__ stage,
                                              const float* __restrict__ sW2,
                                              int lm, int half, v8f acc[4]) {
#pragma unroll
  for (int kt = 0; kt < 16; ++kt) {
    const int k = kt * 4 + half * 2;
    const v2f a = *(const v2f*)(stage + lm * STP + k);
    const float* wp = sW2 + ((k >> 1) << 7);
#pragma unroll
    for (int nt = 0; nt < 4; ++nt) {
      const v2f b = *(const v2f*)(wp + ((nt * 16 + lm) << 1));
      acc[nt] = wmma_f32_16x16x4(a, b, acc[nt]);
    }
  }
}

// -------------------------------------------------------------------------
// Edge kernel: per 16-edge tile compute msg_h, coef, scatter to accumulators
// -------------------------------------------------------------------------
__global__ __launch_bounds__(BLOCK) void egnn_edge_kernel(
    const float* __restrict__ node_feat, const float* __restrict__ coord,
    const int* __restrict__ src, const int* __restrict__ dst,
    const float* __restrict__ We1, const float* __restrict__ be1,
    const float* __restrict__ We2, const float* __restrict__ be2,
    const float* __restrict__ Wx1, const float* __restrict__ bx1,
    const float* __restrict__ Wx2,
    float* __restrict__ h_acc, float* __restrict__ x_acc,
    float* __restrict__ deg, int nE) {
  __shared__ __align__(16) float sW[128 * 64];   // phase-overlaid, pair-major
  __shared__ float sRad[64];                      // We1 row 128 (radial)
  __shared__ float sB[64];
  __shared__ float sWx2[64];
  __shared__ __align__(16) float sStage[WAVES][16 * STP];
  __shared__ float sMeta[WAVES][16 * 4];          // xd0,xd1,xd2,radial
  __shared__ int sSrc[WAVES][16];
  __shared__ int sDst[WAVES][16];
  __shared__ float sCoef[WAVES][16];

  const int tid = threadIdx.x;
  const int wave = tid >> 5;
  const int lane = tid & 31;
  const int half = lane >> 4;
  const int lm = lane & 15;

  float* stage = sStage[wave];
  const long e0 = ((long)blockIdx.x * WAVES + wave) * 16;

  // ---- phase 0: We1 rows 0..127 (pair-major) + radial row + be1 ----
  load_w_pairmajor(We1, sW, 128, tid);
  if (tid < 64) {
    sRad[tid] = We1[128 * 64 + tid];
    sB[tid] = be1[tid];
  }

  if (lane < 16) {
    long e = e0 + lane;
    long ec = e < nE ? e : (long)(nE - 1);
    int s = src[ec], d = dst[ec];
    sSrc[wave][lane] = s;
    sDst[wave][lane] = d;
    float dx = coord[(long)s * 3 + 0] - coord[(long)d * 3 + 0];
    float dy = coord[(long)s * 3 + 1] - coord[(long)d * 3 + 1];
    float dz = coord[(long)s * 3 + 2] - coord[(long)d * 3 + 2];
    float r = dx * dx + dy * dy + dz * dz;
    float inv = 1.0f / (sqrtf(r) + 1e-30f);
    sMeta[wave][lane * 4 + 0] = dx * inv;
    sMeta[wave][lane * 4 + 1] = dy * inv;
    sMeta[wave][lane * 4 + 2] = dz * inv;
    sMeta[wave][lane * 4 + 3] = r;
  }
  __syncthreads();

  v8f acc[4];
#pragma unroll
  for (int nt = 0; nt < 4; ++nt) acc[nt] = v8f_zero();

  // stage h_src tile, GEMM against We1 rows 0..63
  {
    const int node = sSrc[wave][lm];
    const float* gp = node_feat + (long)node * 64 + half * 32;
    float* sp = stage + lm * STP + half * 32;
#pragma unroll
    for (int j = 0; j < 32; j += 4) *(float4*)(sp + j) = *(const float4*)(gp + j);
  }
  __syncthreads();
  gemm_16x64x64(stage, sW, lm, half, acc);
  __syncthreads();

  // stage h_dst tile, GEMM against We1 rows 64..127
  {
    const int node = sDst[wave][lm];
    const float* gp = node_feat + (long)node * 64 + half * 32;
    float* sp = stage + lm * STP + half * 32;
#pragma unroll
    for (int j = 0; j < 32; j += 4) *(float4*)(sp + j) = *(const float4*)(gp + j);
  }
  __syncthreads();
  gemm_16x64x64(stage, sW + 64 * 64, lm, half, acc);

  // rank-1 radial row + bias + SiLU, in C layout
#pragma unroll
  for (int nt = 0; nt < 4; ++nt) {
    const float wr = sRad[nt * 16 + lm];
    const float bn = sB[nt * 16 + lm];
#pragma unroll
    for (int r = 0; r < 8; ++r) {
      const int m = r + half * 8;
      acc[nt][r] = silu_f(acc[nt][r] + sMeta[wave][m * 4 + 3] * wr + bn);
    }
  }
  __syncthreads();
  // write u to stage (C layout -> row major)
#pragma unroll
  for (int nt = 0; nt < 4; ++nt)
#pragma unroll
    for (int r = 0; r < 8; ++r)
      stage[(r + half * 8) * STP + nt * 16 + lm] = acc[nt][r];

  // ---- phase 1: We2 + be2 ----
  __syncthreads();
  load_w_pairmajor(We2, sW, 64, tid);
  if (tid < 64) sB[tid] = be2[tid];
  __syncthreads();

  v8f acc2[4];
#pragma unroll
  for (int nt = 0; nt < 4; ++nt) acc2[nt] = v8f_zero();
  gemm_16x64x64(stage, sW, lm, half, acc2);
#pragma unroll
  for (int nt = 0; nt < 4; ++nt) {
    const float bn = sB[nt * 16 + lm];
#pragma unroll
    for (int r = 0; r < 8; ++r) acc2[nt][r] = silu_f(acc2[nt][r] + bn);
  }
  __syncthreads();
  // msg_h -> stage
#pragma unroll
  for (int nt = 0; nt < 4; ++nt)
#pragma unroll
    for (int r = 0; r < 8; ++r)
      stage[(r + half * 8) * STP + nt * 16 + lm] = acc2[nt][r];
  __syncthreads();

  // scatter msg_h into h_acc (sum reduce)
  {
    const int r = lane >> 1;
    const long e = e0 + r;
    if (e < nE) {
      const int d2 = sDst[wave][r];
      const int cb = (lane & 1) * 32;
      float* hp = h_acc + (long)d2 * 64 + cb;
      const float* spp = stage + r * STP + cb;
#pragma unroll
      for (int c = 0; c < 32; ++c) atomicAdd(hp + c, spp[c]);
    }
  }

  // ---- phase 2: Wx1 + bx1 + Wx2 ----
  __syncthreads();
  load_w_pairmajor(Wx1, sW, 64, tid);
  if (tid < 64) {
    sB[tid] = bx1[tid];
    sWx2[tid] = Wx2[tid];
  }
  if (lane < 16) sCoef[wave][lane] = 0.0f;
  __syncthreads();

  v8f acc3[4];
#pragma unroll
  for (int nt = 0; nt < 4; ++nt) acc3[nt] = v8f_zero();
  gemm_16x64x64(stage, sW, lm, half, acc3);

  float part[8];
#pragma unroll
  for (int r = 0; r < 8; ++r) part[r] = 0.0f;
#pragma unroll
  for (int nt = 0; nt < 4; ++nt) {
    const float bn = sB[nt * 16 + lm];
    const float w2 = sWx2[nt * 16 + lm];
#pragma unroll
    for (int r = 0; r < 8; ++r) part[r] += silu_f(acc3[nt][r] + bn) * w2;
  }
#pragma unroll
  for (int r = 0; r < 8; ++r) atomicAdd(&sCoef[wave][r + half * 8], part[r]);
  __syncthreads();

  if (lane < 16) {
    const long e = e0 + lane;
    if (e < nE) {
      const float coef = sCoef[wave][lane];
      const int d2 = sDst[wave][lane];
      atomicAdd(&x_acc[(long)d2 * 3 + 0], coef * sMeta[wave][lane * 4 + 0]);
      atomicAdd(&x_acc[(long)d2 * 3 + 1], coef * sMeta[wave][lane * 4 + 1]);
      atomicAdd(&x_acc[(long)d2 * 3 + 2], coef * sMeta[wave][lane * 4 + 2]);
      atomicAdd(&deg[d2], 1.0f);
    }
  }
}

// -------------------------------------------------------------------------
// Node kernel: node_mlp on concat(node_feat, h_neigh), plus coord update
// -------------------------------------------------------------------------
__global__ __launch_bounds__(BLOCK) void egnn_node_kernel(
    const float* __restrict__ node_feat, const float* __restrict__ coord,
    const float* __restrict__ Wn1, const float* __restrict__ bn1,
    const float* __restrict__ Wn2, const float* __restrict__ bn2,
    const float* __restrict__ h_acc, const float* __restrict__ x_acc,
    const float* __restrict__ deg,
    float* __restrict__ out_h, float* __restrict__ out_x, int nN) {
  __shared__ __align__(16) float sW[128 * 64];    // pair-major
  __shared__ float sB[64];
  __shared__ __align__(16) float sStage[WAVES][16 * STP];

  const int tid = threadIdx.x;
  const int wave = tid >> 5;
  const int lane = tid & 31;
  const int half = lane >> 4;
  const int lm = lane & 15;

  float* stage = sStage[wave];
  const long n0 = ((long)blockIdx.x * WAVES + wave) * 16;

  load_w_pairmajor(Wn1, sW, 128, tid);
  if (tid < 64) sB[tid] = bn1[tid];
  __syncthreads();

  v8f acc[4];
#pragma unroll
  for (int nt = 0; nt < 4; ++nt) acc[nt] = v8f_zero();

  // stage node_feat rows (clamped)
  {
    long nd = n0 + lm;
    if (nd >= nN) nd = nN - 1;
    const float* gp = node_feat + nd * 64 + half * 32;
    float* sp = stage + lm * STP + half * 32;
#pragma unroll
    for (int j = 0; j < 32; j += 4) *(float4*)(sp + j) = *(const float4*)(gp + j);
  }
  __syncthreads();
  gemm_16x64x64(stage, sW, lm, half, acc);  // Wn1 rows 0..63
  __syncthreads();

  // stage h_neigh rows (clamped)
  {
    long nd = n0 + lm;
    if (nd >= nN) nd = nN - 1;
    const float* gp = h_acc + nd * 64 + half * 32;
    float* sp = stage + lm * STP + half * 32;
#pragma unroll
    for (int j = 0; j < 32; j += 4) *(float4*)(sp + j) = *(const float4*)(gp + j);
  }
  __syncthreads();
  gemm_16x64x64(stage, sW + 64 * 64, lm, half, acc);  // Wn1 rows 64..127

#pragma unroll
  for (int nt = 0; nt < 4; ++nt) {
    const float bn = sB[nt * 16 + lm];
#pragma unroll
    for (int r = 0; r < 8; ++r) acc[nt][r] = silu_f(acc[nt][r] + bn);
  }
  __syncthreads();
#pragma unroll
  for (int nt = 0; nt < 4; ++nt)
#pragma unroll
    for (int r = 0; r < 8; ++r)
      stage[(r + half * 8) * STP + nt * 16 + lm] = acc[nt][r];

  __syncthreads();
  load_w_pairmajor(Wn2, sW, 64, tid);
  if (tid < 64) sB[tid] = bn2[tid];
  __syncthreads();

  v8f acc2[4];
#pragma unroll
  for (int nt = 0; nt < 4; ++nt) acc2[nt] = v8f_zero();
  gemm_16x64x64(stage, sW, lm, half, acc2);

  // h output (+bn2), guarded
#pragma unroll
  for (int nt = 0; nt < 4; ++nt) {
    const float bn = sB[nt * 16 + lm];
#pragma unroll
    for (int r = 0; r < 8; ++r) {
      const int m = r + half * 8;
      const long nd = n0 + m;
      if (nd < nN) out_h[nd * 64 + nt * 16 + lm] = acc2[nt][r] + bn;
    }
  }

  // x output: coord + x_sum / max(deg, 1)
  if (lane < 16) {
    const long nd = n0 + lane;
    if (nd < nN) {
      float dg = deg[nd];
      dg = dg > 1.0f ? dg : 1.0f;
      const float rdg = 1.0f / dg;
      out_x[nd * 3 + 0] = coord[nd * 3 + 0] + x_acc[nd * 3 + 0] * rdg;
      out_x[nd * 3 + 1] = coord[nd * 3 + 1] + x_acc[nd * 3 + 1] * rdg;
      out_x[nd * 3 + 2] = coord[nd * 3 + 2] + x_acc[nd * 3 + 2] * rdg;
    }
  }
}

extern "C" void kernel_launch(void* const* d_in, const int* in_sizes, int n_in,
                              void* d_out, int out_size, void* d_ws,
                              size_t ws_size, hipStream_t stream) {
  const float* node_feat = (const float*)d_in[0];
  const float* coord = (const float*)d_in[1];
  const int* src = (const int*)d_in[2];
  const int* dst = (const int*)d_in[3];
  const float* We1 = (const float*)d_in[4];
  const float* be1 = (const float*)d_in[5];
  const float* We2 = (const float*)d_in[6];
  const float* be2 = (const float*)d_in[7];
  const float* Wx1 = (const float*)d_in[8];
  const float* bx1 = (const float*)d_in[9];
  const float* Wx2 = (const float*)d_in[10];
  const float* Wn1 = (const float*)d_in[11];
  const float* bn1 = (const float*)d_in[12];
  const float* Wn2 = (const float*)d_in[13];
  const float* bn2 = (const float*)d_in[14];

  const int nN = in_sizes[0] / 64;
  const int nE = in_sizes[2];

  // workspace accumulators: h_neigh [N,64] | x_sum [N,3] | deg [N]
  float* h_acc = (float*)d_ws;
  float* x_acc = h_acc + (size_t)nN * 64;
  float* deg = x_acc + (size_t)nN * 3;
  hipMemsetAsync(d_ws, 0, (size_t)nN * 68 * sizeof(float), stream);

  const int edge_tiles = (nE + 15) / 16;
  const int edge_blocks = (edge_tiles + WAVES - 1) / WAVES;
  egnn_edge_kernel<<<edge_blocks, BLOCK, 0, stream>>>(
      node_feat, coord, src, dst, We1, be1, We2, be2, Wx1, bx1, Wx2, h_acc,
      x_acc, deg, nE);

  float* out_h = (float*)d_out;
  float* out_x = out_h + (size_t)nN * 64;
  const int node_tiles = (nN + 15) / 16;
  const int node_blocks = (node_tiles + WAVES - 1) / WAVES;
  egnn_node_kernel<<<node_blocks, BLOCK, 0, stream>>>(
      node_feat, coord, Wn1, bn1, Wn2, bn2, h_acc, x_acc, deg, out_h, out_x,
      nN);
}